// GaussianKernel_24773371363550
// MI455X (gfx1250) — compile-verified
//
#include <hip/hip_runtime.h>
#include <math.h>

typedef __attribute__((ext_vector_type(2))) float v2f;
typedef __attribute__((ext_vector_type(8))) float v8f;

#define DIM     512
#define CHUNKS  32      // train-column chunks; partials per test row
#define WAVES   8       // waves per block (256 threads, wave32)

// ---------------------------------------------------------------------------
// Kernel 0: squared row norms of test (x2) and train (y2). One wave per row.
// ---------------------------------------------------------------------------
__global__ __launch_bounds__(256) void kde_norms_kernel(
    const float* __restrict__ test, const float* __restrict__ train,
    float* __restrict__ x2, float* __restrict__ y2, int n_test, int n_train) {
  const int wave = threadIdx.x >> 5;
  const int lane = threadIdx.x & 31;
  const int row  = blockIdx.x * WAVES + wave;
  const int total = n_test + n_train;
  if (row >= total) return;
  const float* src;
  float* dst;
  if (row < n_test) { src = test + (size_t)row * DIM;             dst = x2 + row; }
  else              { src = train + (size_t)(row - n_test) * DIM; dst = y2 + (row - n_test); }
  float s = 0.0f;
  for (int i = lane; i < DIM; i += 32) { float v = src[i]; s = fmaf(v, v, s); }
  #pragma unroll
  for (int m = 16; m >= 1; m >>= 1) s += __shfl_xor(s, m, 32);
  if (lane == 0) *dst = s;
}

// ---------------------------------------------------------------------------
// Kernel 1: WMMA f32 16x16x4, dual-tile per wave (one A fragment feeds two
// independent WMMA chains), software-pipelined K loop with RUNTIME trip count
// (prevents full unroll -> keeps VGPRs < 256, no s_set_vgpr_msb churn),
// per-lane online logsumexp accumulation (no cross-lane ops in hot path).
//   blockIdx.y -> test-row tile (16 rows), blockIdx.x -> train-col chunk (512).
// ---------------------------------------------------------------------------
__global__ __launch_bounds__(256) void kde_tile_kernel(
    const float* __restrict__ test, const float* __restrict__ train,
    const float* __restrict__ x2,   const float* __restrict__ y2,
    float2* __restrict__ part, int n_train, int dim) {
  const int lane = threadIdx.x & 31;
  const int wave = threadIdx.x >> 5;
  const int hi   = lane >> 4;       // 0: lanes 0-15, 1: lanes 16-31
  const int lo   = lane & 15;
  const int trow0 = blockIdx.y * 16;
  const int chunk = blockIdx.x;
  const int chunk_cols = n_train / CHUNKS;          // 512
  const int pairs = chunk_cols / (WAVES * 32);      // 2
  const int chunk_base = chunk * chunk_cols;
  const int koff = hi * 2;          // A/B fragment K offset for this half-wave

  __shared__ float2 lds[WAVES][16];

  // x2 values for the rows this lane's accumulator VGPRs correspond to
  float x2v[8];
  #pragma unroll
  for (int v = 0; v < 8; ++v) x2v[v] = x2[trow0 + v + hi * 8];

  const float* Abase = test + (size_t)(trow0 + lo) * dim + koff;

  // per-lane online (max, sumexp) accumulators, one per row of the tile
  float acc_m[8], acc_s[8];
  #pragma unroll
  for (int v = 0; v < 8; ++v) { acc_m[v] = -INFINITY; acc_s[v] = 0.0f; }

  for (int t = 0; t < pairs; ++t) {
    const int col0 = chunk_base + (t * WAVES + wave) * 32;
    const float* B0 = train + (size_t)(col0 + lo) * dim + koff;   // tile 0
    const float* B1 = B0 + (size_t)16 * dim;                      // tile 1

    v8f c0 = {}, c1 = {};
    // software pipeline: next-step fragments stay in flight across the WMMAs
    v2f a  = *(const v2f*)(Abase);
    v2f b0 = *(const v2f*)(B0);
    v2f b1 = *(const v2f*)(B1);
    const int kend = dim - 4;       // runtime bound: no full unroll
    #pragma unroll 4
    for (int k = 0; k < kend; k += 4) {
      v2f an  = *(const v2f*)(Abase + k + 4);
      v2f b0n = *(const v2f*)(B0 + k + 4);
      v2f b1n = *(const v2f*)(B1 + k + 4);
      c0 = __builtin_amdgcn_wmma_f32_16x16x4_f32(false, a, false, b0,
                                                 (short)0, c0, false, false);
      c1 = __builtin_amdgcn_wmma_f32_16x16x4_f32(false, a, false, b1,
                                                 (short)0, c1, false, false);
      a = an; b0 = b0n; b1 = b1n;
    }
    c0 = __builtin_amdgcn_wmma_f32_16x16x4_f32(false, a, false, b0,
                                               (short)0, c0, false, false);
    c1 = __builtin_amdgcn_wmma_f32_16x16x4_f32(false, a, false, b1,
                                               (short)0, c1, false, false);

    // epilogue: c{0,1}[v] -> log-kernel values; fold into per-lane online pair
    const float y20 = y2[col0 + lo];
    const float y21 = y2[col0 + 16 + lo];
    #pragma unroll
    for (int v = 0; v < 8; ++v) {
      float le0 = -0.5f * fmaxf(x2v[v] + y20 - 2.0f * c0[v], 0.0f);
      float le1 = -0.5f * fmaxf(x2v[v] + y21 - 2.0f * c1[v], 0.0f);
      float m   = fmaxf(le0, le1);
      float s   = __expf(le0 - m) + __expf(le1 - m);
      float m2  = fmaxf(acc_m[v], m);
      acc_s[v]  = acc_s[v] * __expf(acc_m[v] - m2) + s * __expf(m - m2);
      acc_m[v]  = m2;
    }
  }

  // one cross-lane reduction at the end: combine (max,sum) pairs across the
  // 16 lanes of each half-wave (xor masks stay within the halves)
  #pragma unroll
  for (int v = 0; v < 8; ++v) {
    float mx = acc_m[v], s = acc_s[v];
    #pragma unroll
    for (int m = 1; m <= 8; m <<= 1) {
      float omx = __shfl_xor(mx, m, 32);
      float os  = __shfl_xor(s,  m, 32);
      float m2  = fmaxf(mx, omx);
      s  = s * __expf(mx - m2) + os * __expf(omx - m2);
      mx = m2;
    }
    acc_m[v] = mx; acc_s[v] = s;
  }

  if (lo == 0) {
    #pragma unroll
    for (int v = 0; v < 8; ++v)
      lds[wave][v + hi * 8] = make_float2(acc_m[v], acc_s[v]);
  }
  __syncthreads();

  // combine the 8 waves' pairs; one thread per test row of the tile
  if (threadIdx.x < 16) {
    const int r = threadIdx.x;
    float mx = -INFINITY, s = 0.0f;
    #pragma unroll
    for (int w = 0; w < WAVES; ++w) {
      float2 p = lds[w][r];
      float m2 = fmaxf(mx, p.x);
      s = s * __expf(mx - m2) + p.y * __expf(p.x - m2);
      mx = m2;
    }
    part[(size_t)(trow0 + r) * CHUNKS + chunk] = make_float2(mx, s);
  }
}

// ---------------------------------------------------------------------------
// Kernel 2: merge the 32 chunk partials per row (online logsumexp), finalize.
// ---------------------------------------------------------------------------
__global__ __launch_bounds__(32) void kde_reduce_kernel(
    const float2* __restrict__ part, float* __restrict__ out, float z) {
  const int row  = blockIdx.x;
  const int lane = threadIdx.x;
  float2 p = part[(size_t)row * CHUNKS + lane];
  float mx = p.x, s = p.y;
  #pragma unroll
  for (int m = 16; m >= 1; m >>= 1) {
    float omx = __shfl_xor(mx, m, 32);
    float os  = __shfl_xor(s,  m, 32);
    float m2  = fmaxf(mx, omx);
    s  = s * __expf(mx - m2) + os * __expf(omx - m2);
    mx = m2;
  }
  if (lane == 0) out[row] = mx + logf(s) - z;
}

// ---------------------------------------------------------------------------
extern "C" void kernel_launch(void* const* d_in, const int* in_sizes, int n_in,
                              void* d_out, int out_size, void* d_ws, size_t ws_size,
                              hipStream_t stream) {
  const float* test  = (const float*)d_in[0];
  const float* train = (const float*)d_in[1];
  const int n_test  = in_sizes[0] / DIM;   // 8192
  const int n_train = in_sizes[1] / DIM;   // 16384

  float* ws_f = (float*)d_ws;
  float* x2 = ws_f;
  float* y2 = ws_f + n_test;
  float2* part = (float2*)(ws_f + n_test + n_train);  // 8-byte aligned offset
  float* out = (float*)d_out;

  const int rows = n_test + n_train;
  kde_norms_kernel<<<(rows + WAVES - 1) / WAVES, 256, 0, stream>>>(
      test, train, x2, y2, n_test, n_train);

  dim3 grid(CHUNKS, n_test / 16);
  kde_tile_kernel<<<grid, 256, 0, stream>>>(test, train, x2, y2, part,
                                            n_train, DIM);

  // z = 0.5*d*log(2*pi) + d*log(bw) + log(n_train); bw = 1
  const float z = 0.5f * (float)DIM * logf(6.283185307179586f) + logf((float)n_train);
  kde_reduce_kernel<<<n_test, 32, 0, stream>>>(part, out, z);
}